// flow_model_73950746902692
// MI455X (gfx1250) — compile-verified
//
#include <hip/hip_runtime.h>
#include <math.h>

// CDNA5 / gfx1250. wave32. f32 WMMA (V_WMMA_F32_16X16X4_F32) for the dominant
// (64x256)x(256x128) GEMM; everything else is latency-trivial elementwise.

typedef __attribute__((ext_vector_type(2))) float v2f;
typedef __attribute__((ext_vector_type(8))) float v8f;

#define FEAT 256   // F
#define NB   64    // batch
#define N1   128   // fc1 width

// ---------------------------------------------------------------------------
// Kernel A: gather x = flow[b, 63, 32, 32, :]; GRU gates with h_prev == 0
// (so W_hh is dead, gh == b_hh).  One block per batch, one thread per feature.
// ---------------------------------------------------------------------------
__global__ __launch_bounds__(FEAT) void flow_gates_kernel(
    const float* __restrict__ flow, const float* __restrict__ W_ih,
    const float* __restrict__ b_ih, const float* __restrict__ b_hh,
    float* __restrict__ h)
{
    const int b = blockIdx.x;
    const int f = threadIdx.x;
    // flow shape (64,64,64,64,2); index [b, 63, 32, 32, :]
    const size_t base = (size_t)b * (64ull * 64 * 64 * 2)
                      + 63ull * (64 * 64 * 2)
                      + 32ull * (64 * 2)
                      + 32ull * 2;
    const float x0 = flow[base + 0];
    const float x1 = flow[base + 1];

    // gx[j] = x0*W_ih[j][0] + x1*W_ih[j][1] + b_ih[j],  W_ih row-major (768,2)
    const int jr = 0 * FEAT + f, jz = 1 * FEAT + f, jn = 2 * FEAT + f;
    const float rx = fmaf(x0, W_ih[jr * 2 + 0], fmaf(x1, W_ih[jr * 2 + 1], b_ih[jr]));
    const float zx = fmaf(x0, W_ih[jz * 2 + 0], fmaf(x1, W_ih[jz * 2 + 1], b_ih[jz]));
    const float nx = fmaf(x0, W_ih[jn * 2 + 0], fmaf(x1, W_ih[jn * 2 + 1], b_ih[jn]));

    const float r = 1.0f / (1.0f + __expf(-(rx + b_hh[jr])));
    const float z = 1.0f / (1.0f + __expf(-(zx + b_hh[jz])));
    const float n = tanhf(fmaf(r, b_hh[jn], nx));
    h[b * FEAT + f] = (1.0f - z) * n;
}

// ---------------------------------------------------------------------------
// Kernel B: y1 = relu(h @ W1^T + b1) via V_WMMA_F32_16X16X4_F32.
// h: (64,256) row-major. W1: (128,256) row-major, so B[k][n] = W1[n][k] and
// each lane's B column fragment is a contiguous float2 along K.
// One wave32 per 16x16 output tile; 32 tiles; K-loop of 64 steps (K=4 each).
//
// A frag (16x4 f32): lane L<16 -> M=L; VGPR j, lane-half g -> K = 2*g + j.
// B frag (4x16 f32): lane L<16 -> N=L; same K striping (symmetric layout).
// D (16x16 f32, 8 VGPRs): vgpr i, lanes 0-15 -> M=i; lanes 16-31 -> M=i+8.
// ---------------------------------------------------------------------------
__global__ __launch_bounds__(32) void fc1_wmma_kernel(
    const float* __restrict__ h, const float* __restrict__ W1,
    const float* __restrict__ b1, float* __restrict__ y1)
{
    const int tile = blockIdx.x;          // 0..31
    const int tm = (tile >> 3) * 16;      // M tile base: 0,16,32,48
    const int tn = (tile & 7) * 16;       // N tile base: 0..112
    const int lane = threadIdx.x;         // wave32, EXEC all ones (no divergence)
    const int l15  = lane & 15;
    const int koff = (lane >> 4) * 2;     // lane-half K offset

    const float* arow = h  + (size_t)(tm + l15) * FEAT;  // A row  (M = tm+l15)
    const float* brow = W1 + (size_t)(tn + l15) * FEAT;  // B col  (N = tn+l15)

    v8f acc = {};
    #pragma unroll 8
    for (int k = 0; k < FEAT; k += 4) {
        const v2f a = *(const v2f*)(arow + k + koff);
        const v2f b = *(const v2f*)(brow + k + koff);
        // (neg_a, A, neg_b, B, c_mod, C, reuse_a, reuse_b)
        acc = __builtin_amdgcn_wmma_f32_16x16x4_f32(
            false, a, false, b, (short)0, acc, false, false);
    }

    const float bias = b1[tn + l15];
    const int rbase = tm + ((lane >> 4) << 3);  // +8 rows for upper lane-half
    #pragma unroll
    for (int i = 0; i < 8; ++i) {
        const float v = acc[i] + bias;
        y1[(size_t)(rbase + i) * N1 + (tn + l15)] = v > 0.0f ? v : 0.0f;
    }
}

// ---------------------------------------------------------------------------
// Kernel C: y2 = tanh(y1 @ W2^T + b2)  (64x128 . 128x2), then
//           out = y2 @ W3^T + b3       (64x2 . 2x2).  One thread per batch.
// ---------------------------------------------------------------------------
__global__ __launch_bounds__(NB) void head_kernel(
    const float* __restrict__ y1, const float* __restrict__ W2,
    const float* __restrict__ b2, const float* __restrict__ W3,
    const float* __restrict__ b3, float* __restrict__ out)
{
    const int b = threadIdx.x;
    const float* row = y1 + b * N1;
    float a0 = b2[0], a1 = b2[1];
    #pragma unroll 4
    for (int i = 0; i < N1; ++i) {
        const float v = row[i];
        a0 = fmaf(v, W2[i], a0);          // W2 row 0
        a1 = fmaf(v, W2[N1 + i], a1);     // W2 row 1
    }
    const float t0 = tanhf(a0);
    const float t1 = tanhf(a1);
    // out[j] = t0*W3[j][0] + t1*W3[j][1] + b3[j],  W3 row-major (2,2)
    out[b * 2 + 0] = fmaf(t0, W3[0], fmaf(t1, W3[1], b3[0]));
    out[b * 2 + 1] = fmaf(t0, W3[2], fmaf(t1, W3[3], b3[1]));
}

// ---------------------------------------------------------------------------
extern "C" void kernel_launch(void* const* d_in, const int* in_sizes, int n_in,
                              void* d_out, int out_size, void* d_ws, size_t ws_size,
                              hipStream_t stream) {
    const float* flow = (const float*)d_in[0];
    const float* W_ih = (const float*)d_in[1];
    const float* b_ih = (const float*)d_in[2];
    // d_in[3] = W_hh : dead (h_prev == 0 in the reference)
    const float* b_hh = (const float*)d_in[4];
    const float* W1   = (const float*)d_in[5];
    const float* b1   = (const float*)d_in[6];
    const float* W2   = (const float*)d_in[7];
    const float* b2   = (const float*)d_in[8];
    const float* W3   = (const float*)d_in[9];
    const float* b3   = (const float*)d_in[10];
    float* out = (float*)d_out;

    float* h  = (float*)d_ws;          // 64*256 f32 = 64 KB
    float* y1 = h + NB * FEAT;         // 64*128 f32 = 32 KB

    flow_gates_kernel<<<NB, FEAT, 0, stream>>>(flow, W_ih, b_ih, b_hh, h);
    fc1_wmma_kernel<<<32, 32, 0, stream>>>(h, W1, b1, y1);
    head_kernel<<<1, NB, 0, stream>>>(y1, W2, b2, W3, b3, out);
}